// TreeModelLogReg_11673721110709
// MI455X (gfx1250) — compile-verified
//
#include <hip/hip_runtime.h>
#include <hip/hip_bf16.h>
#include <math.h>

#define DEPTH     15
#define N_LEAVES  32768
#define N_NODES   65535
#define FDIM      256
#define CDIM      2
#define NODE_ELEMS (FDIM * CDIM)   // 512 floats per node
#define MIN_DIST  1e-7f

typedef __attribute__((ext_vector_type(2))) float v2f;
typedef __attribute__((ext_vector_type(8))) float v8f;

// ---------------------------------------------------------------------------
// Kernel 1: per-wave partial sums of the delta regularizer.
// One wave handles one node per iteration (512 floats = 4x float4 per lane,
// coalesced b128). Fixed node->wave assignment + fixed partial slots =>
// bitwise deterministic. Prefetch hides latency on the 134 MB delta stream.
// ---------------------------------------------------------------------------
#define REG_BLOCKS 1024
#define REG_TPB    256
#define REG_WAVES  (REG_TPB / 32)
#define REG_NWAVES (REG_BLOCKS * REG_WAVES)   // 8192 partials

__global__ __launch_bounds__(REG_TPB)
void reg_partials_kernel(const float* __restrict__ deltas,
                         const float* __restrict__ branch,
                         float* __restrict__ partials)
{
    const int lane  = threadIdx.x & 31;
    const int wave  = threadIdx.x >> 5;
    const int gwave = blockIdx.x * REG_WAVES + wave;

    float accum = 0.0f;
    for (int n = gwave; n < N_NODES; n += REG_NWAVES) {
        if (n + REG_NWAVES < N_NODES) {
            __builtin_prefetch(deltas + (size_t)(n + REG_NWAVES) * NODE_ELEMS + lane * 16, 0, 0);
        }
        const float4* dp = (const float4*)(deltas + (size_t)n * NODE_ELEMS) + lane * 4;
        const float4 a0 = dp[0], a1 = dp[1], a2 = dp[2], a3 = dp[3];
        float s;
        if (n == 0) {
            s = a0.x*a0.x + a0.y*a0.y + a0.z*a0.z + a0.w*a0.w
              + a1.x*a1.x + a1.y*a1.y + a1.z*a1.z + a1.w*a1.w
              + a2.x*a2.x + a2.y*a2.y + a2.z*a2.z + a2.w*a2.w
              + a3.x*a3.x + a3.y*a3.y + a3.z*a3.z + a3.w*a3.w;
        } else {
            s = fabsf(a0.x)+fabsf(a0.y)+fabsf(a0.z)+fabsf(a0.w)
              + fabsf(a1.x)+fabsf(a1.y)+fabsf(a1.z)+fabsf(a1.w)
              + fabsf(a2.x)+fabsf(a2.y)+fabsf(a2.z)+fabsf(a2.w)
              + fabsf(a3.x)+fabsf(a3.y)+fabsf(a3.z)+fabsf(a3.w);
        }
        // full wave32 xor-reduction: every lane ends with the node sum
        #pragma unroll
        for (int off = 16; off >= 1; off >>= 1) s += __shfl_xor(s, off, 32);
        if (n != 0) s /= fmaxf(branch[n - 1], MIN_DIST);
        accum += s;
    }
    if (lane == 0) partials[gwave] = accum;
}

// ---------------------------------------------------------------------------
// Kernel 2: deterministic single-block tree reduction of the 8192 partials.
// ---------------------------------------------------------------------------
__global__ __launch_bounds__(1024)
void reg_reduce_kernel(const float* __restrict__ partials, float* __restrict__ out)
{
    __shared__ float sm[1024];
    float s = 0.0f;
    for (int i = threadIdx.x; i < REG_NWAVES; i += 1024) s += partials[i];
    sm[threadIdx.x] = s;
    __syncthreads();
    for (int w = 512; w >= 1; w >>= 1) {
        if ((int)threadIdx.x < w) sm[threadIdx.x] += sm[threadIdx.x + w];
        __syncthreads();
    }
    if (threadIdx.x == 0) out[2 * N_LEAVES] = sm[0];   // delta_tensor scalar
}

// ---------------------------------------------------------------------------
// Kernel 3: logits + softmax. One wave owns a tile of 16 consecutive leaves.
//
// Shared ancestors (levels 0..11, identical for all 16 leaves of a tile)
// accumulate into W_shared[256][2]; their logit contribution is computed as
//    C[2(pad16),16] = W^T[2(pad16),256] x X^T[256,16]
// via 64x v_wmma_f32_16x16x4_f32. Rows M>=2 of A carry duplicated (finite)
// data instead of zeros -> unconditional fragment loads, no EXEC masking;
// those rows of C are simply never read. cfrag[0]/cfrag[1] in lanes 0..15
// directly hold logits(leaf=lane, c=0/1).
//
// Residual levels 12..15 (per-leaf weights) are wave-cooperative dot
// products with 128-bit loads; they dominate bytes and are not a GEMM.
// ---------------------------------------------------------------------------
#define LG_WAVES 2                  // waves (tiles) per block
#define XS_STRIDE 260               // floats; 16B-aligned rows, conflict-free columns

__global__ __launch_bounds__(LG_WAVES * 32)
void logits_softmax_kernel(const float* __restrict__ x,
                           const float* __restrict__ deltas,
                           float* __restrict__ out)
{
    __shared__ float xs[LG_WAVES][16 * XS_STRIDE];   // X tile, row-major (padded)
    __shared__ float wsh[LG_WAVES][NODE_ELEMS];      // W_shared, layout [f*2 + c]
    __shared__ float lg[LG_WAVES][16 * CDIM];        // residual logit accumulators

    const int lane  = threadIdx.x & 31;
    const int wave  = threadIdx.x >> 5;
    const int tile  = blockIdx.x * LG_WAVES + wave;  // grid sized exactly
    const int leaf0 = tile * 16;

    float* xw  = xs[wave];
    float* ww  = wsh[wave];
    float* lgw = lg[wave];

    // --- stage X tile into LDS: 1024 float4, 32 per lane, ds_store_b128 ---
    {
        const float4* xb4 = (const float4*)(x + (size_t)leaf0 * FDIM);
        for (int t = lane; t < (16 * FDIM) / 4; t += 32) {
            const int r = t >> 6, c4 = t & 63;
            *(float4*)&xw[r * XS_STRIDE + c4 * 4] = xb4[t];
        }
    }

    // --- accumulate shared ancestor weights, levels 0..11 (16 floats/lane) ---
    {
        float4 acc0 = {0,0,0,0}, acc1 = {0,0,0,0}, acc2 = {0,0,0,0}, acc3 = {0,0,0,0};
        #pragma unroll
        for (int l = 0; l <= 11; ++l) {
            const int idx = ((1 << l) - 1) + (leaf0 >> (DEPTH - l));
            const float4* dp = (const float4*)(deltas + (size_t)idx * NODE_ELEMS) + lane * 4;
            const float4 d0 = dp[0], d1 = dp[1], d2 = dp[2], d3 = dp[3];
            acc0.x += d0.x; acc0.y += d0.y; acc0.z += d0.z; acc0.w += d0.w;
            acc1.x += d1.x; acc1.y += d1.y; acc1.z += d1.z; acc1.w += d1.w;
            acc2.x += d2.x; acc2.y += d2.y; acc2.z += d2.z; acc2.w += d2.w;
            acc3.x += d3.x; acc3.y += d3.y; acc3.z += d3.z; acc3.w += d3.w;
        }
        float4* wv = (float4*)(ww) + lane * 4;
        wv[0] = acc0; wv[1] = acc1; wv[2] = acc2; wv[3] = acc3;
    }
    if (lane < 16) { lgw[lane * CDIM + 0] = 0.0f; lgw[lane * CDIM + 1] = 0.0f; }
    __syncthreads();

    // --- WMMA GEMM: C = W^T(2pad16 x 256) x X^T(256 x 16), K in steps of 4 ---
    // A (16x4 f32): lane M=lane&15, VGPR0 = K=k0+koff, VGPR1 = K=k0+koff+1
    // B (4x16 f32): lane N=lane&15, same K split. Rows M>=2 duplicate row (M&1).
    v8f cfrag = {};
    const int mn   = lane & 15;
    const int koff = (lane < 16) ? 0 : 2;
    const int wcol = mn & 1;                 // class column this lane mirrors
    for (int k0 = 0; k0 < FDIM; k0 += 4) {
        const int k = k0 + koff;
        v2f a, b;
        a.x = ww[k * CDIM + wcol];
        a.y = ww[(k + 1) * CDIM + wcol];
        b.x = xw[mn * XS_STRIDE + k];
        b.y = xw[mn * XS_STRIDE + k + 1];
        cfrag = __builtin_amdgcn_wmma_f32_16x16x4_f32(
            /*neg_a=*/false, a, /*neg_b=*/false, b,
            /*c_mod=*/(short)0, cfrag, /*reuse_a=*/false, /*reuse_b=*/false);
    }

    // --- residual levels 12..15: per-leaf wave-cooperative dot products ---
    for (int j = 0; j < 16; ++j) {
        const int leaf = leaf0 + j;
        const float4* xr4 = (const float4*)(xw + j * XS_STRIDE) + lane * 2;
        const float4 xa = xr4[0], xb = xr4[1];
        float s0 = 0.0f, s1 = 0.0f;
        #pragma unroll
        for (int l = 12; l <= 15; ++l) {
            const int idx = ((1 << l) - 1) + (leaf >> (DEPTH - l));
            const float4* dp = (const float4*)(deltas + (size_t)idx * NODE_ELEMS) + lane * 4;
            const float4 d0 = dp[0], d1 = dp[1], d2 = dp[2], d3 = dp[3];
            // d0 = (f0:c0,c1),(f0+1:c0,c1) ... with f0 = lane*8
            s0 += xa.x*d0.x + xa.y*d0.z + xa.z*d1.x + xa.w*d1.z;
            s1 += xa.x*d0.y + xa.y*d0.w + xa.z*d1.y + xa.w*d1.w;
            s0 += xb.x*d2.x + xb.y*d2.z + xb.z*d3.x + xb.w*d3.z;
            s1 += xb.x*d2.y + xb.y*d2.w + xb.z*d3.y + xb.w*d3.w;
        }
        #pragma unroll
        for (int off = 16; off >= 1; off >>= 1) {
            s0 += __shfl_xor(s0, off, 32);
            s1 += __shfl_xor(s1, off, 32);
        }
        if (lane == 0) {
            lgw[j * CDIM + 0] += s0;
            lgw[j * CDIM + 1] += s1;
        }
    }
    __syncthreads();

    // --- softmax over C=2; lanes 0..15 own leaf = leaf0 + lane ---
    // cfrag[0] = D[M=0][N=lane] = shared logit c=0; cfrag[1] = c=1.
    if (lane < 16) {
        const float z0 = cfrag[0] + lgw[lane * CDIM + 0];
        const float z1 = cfrag[1] + lgw[lane * CDIM + 1];
        const float m  = fmaxf(z0, z1);
        const float e0 = expf(z0 - m);
        const float e1 = expf(z1 - m);
        const float inv = 1.0f / (e0 + e1);
        const size_t o = (size_t)(leaf0 + lane) * CDIM;
        out[o + 0] = e0 * inv;
        out[o + 1] = e1 * inv;
    }
}

// ---------------------------------------------------------------------------
extern "C" void kernel_launch(void* const* d_in, const int* in_sizes, int n_in,
                              void* d_out, int out_size, void* d_ws, size_t ws_size,
                              hipStream_t stream) {
    const float* x      = (const float*)d_in[0];   // [32768, 256]
    const float* deltas = (const float*)d_in[1];   // [65535, 256, 2]
    const float* branch = (const float*)d_in[2];   // [65534]
    float* out      = (float*)d_out;               // 65536 softmax + 1 reg scalar
    float* partials = (float*)d_ws;                // 8192 floats (32 KB)

    reg_partials_kernel<<<REG_BLOCKS, REG_TPB, 0, stream>>>(deltas, branch, partials);
    reg_reduce_kernel<<<1, 1024, 0, stream>>>(partials, out);

    const int n_tiles = N_LEAVES / 16;             // 2048
    logits_softmax_kernel<<<n_tiles / LG_WAVES, LG_WAVES * 32, 0, stream>>>(x, deltas, out);
}